// PGCN_10625749090655
// MI455X (gfx1250) — compile-verified
//
#include <hip/hip_runtime.h>

#define N_NODES 100000
#define N_EDGES 1600000
#define D 128           // D_IN == D_OUT == 128
#define M_TILES (N_NODES / 16)   // 6250, exact

typedef __attribute__((ext_vector_type(2))) float v2f;
typedef __attribute__((ext_vector_type(8))) float v8f;

// ---------------------------------------------------------------- zero AH ---
__global__ void __launch_bounds__(256) pgcn_zero(float4* __restrict__ p, int n4) {
  int i = blockIdx.x * blockDim.x + threadIdx.x;
  if (i < n4) p[i] = make_float4(0.f, 0.f, 0.f, 0.f);
}

// ------------------------------------------------- edge scatter (SpMM) ------
// One wave32 per edge: lane l loads 16B of H[cols[e]] (coalesced 512B/row)
// and does 4 f32 atomic adds into AH[rows[e]]. H and AH are L2-resident.
__global__ void __launch_bounds__(256) pgcn_scatter(
    const float* __restrict__ H, const float* __restrict__ vals,
    const int* __restrict__ rows, const int* __restrict__ cols,
    float* __restrict__ AH) {
  int e = (int)((blockIdx.x * (unsigned)blockDim.x + threadIdx.x) >> 5);
  int lane = threadIdx.x & 31;
  if (e >= N_EDGES) return;
  int r = rows[e];
  int c = cols[e];
  float v = vals[e];
  const float4 h = *(const float4*)(H + (size_t)c * D + lane * 4);
  float* dst = AH + (size_t)r * D + lane * 4;
  atomicAdd(dst + 0, v * h.x);
  atomicAdd(dst + 1, v * h.y);
  atomicAdd(dst + 2, v * h.z);
  atomicAdd(dst + 3, v * h.w);
}

// ------------------------------------- out = relu(AH @ W^T), in-place -------
// out[i][j] = relu(sum_k AH[i][k] * W[j][k]).
// 4 waves / block; each wave owns one 16-row M tile and the full 128-wide N.
// W (64KB) staged in LDS. f32 WMMA 16x16x4, 32 K-steps, 8 N-tiles.
__global__ void __launch_bounds__(128) pgcn_gemm_relu(
    const float* __restrict__ W, float* __restrict__ OUT) {
  __shared__ float sW[D * D];                     // 64 KB
  const int tid = threadIdx.x;

  {   // cooperative 128-thread load of W into LDS (float4, coalesced)
    const float4* W4 = (const float4*)W;
    float4* s4 = (float4*)sW;
#pragma unroll
    for (int i = 0; i < (D * D / 4) / 128; ++i)
      s4[tid + 128 * i] = W4[tid + 128 * i];
  }
  __syncthreads();

  const int wave = tid >> 5;
  const int lane = tid & 31;
  const int tile = blockIdx.x * 4 + wave;         // 16-row M tile
  if (tile >= M_TILES) return;                    // whole-wave exit: EXEC stays all-1 for WMMA

  const int r  = lane & 15;                       // row within A-frag / col within B-frag
  const int hi = lane >> 4;                       // K sub-pair select (0 -> K{0,1}, 1 -> K{2,3})
  const int m0 = tile * 16;

  const float* Arow = OUT + (size_t)(m0 + r) * D + 2 * hi;   // AH lives in OUT
  const float* Brow = sW + (size_t)r * D + 2 * hi;           // W[j][k], j = n0 + r

  v8f acc[8] = {};                                // 8 N-tiles of 16x16 f32 accum

#pragma unroll 8
  for (int kk = 0; kk < 32; ++kk) {               // K = 128 = 32 * 4
    v2f a = *(const v2f*)(Arow + 4 * kk);         // global_load_b64 (L2-hot)
#pragma unroll
    for (int nt = 0; nt < 8; ++nt) {
      v2f b = *(const v2f*)(Brow + nt * 16 * D + 4 * kk);    // ds_load_b64
      acc[nt] = __builtin_amdgcn_wmma_f32_16x16x4_f32(
          false, a, false, b, (short)0, acc[nt], false, false);
    }
  }

  // C/D layout: VGPR g holds row m0+g (lanes 0-15) / m0+8+g (lanes 16-31), N = lane&15.
#pragma unroll
  for (int nt = 0; nt < 8; ++nt) {
#pragma unroll
    for (int g = 0; g < 8; ++g) {
      int m = m0 + g + 8 * hi;
      OUT[(size_t)m * D + nt * 16 + r] = fmaxf(acc[nt][g], 0.f);
    }
  }
}

// ---------------------------------------------------------------------------
extern "C" void kernel_launch(void* const* d_in, const int* in_sizes, int n_in,
                              void* d_out, int out_size, void* d_ws, size_t ws_size,
                              hipStream_t stream) {
  (void)in_sizes; (void)n_in; (void)out_size; (void)d_ws; (void)ws_size;
  const float* H    = (const float*)d_in[0];
  const float* vals = (const float*)d_in[1];
  const float* W    = (const float*)d_in[2];
  const int*   rows = (const int*)d_in[3];
  const int*   cols = (const int*)d_in[4];
  float* out = (float*)d_out;                     // AH accumulates here, then in-place GEMM

  // 1) AH = 0
  const int n4 = N_NODES * D / 4;                 // 3,200,000 float4
  pgcn_zero<<<(n4 + 255) / 256, 256, 0, stream>>>((float4*)out, n4);

  // 2) AH += scatter(vals * H[cols]) ; one wave per edge, 8 edges/block
  pgcn_scatter<<<N_EDGES / 8, 256, 0, stream>>>(H, vals, rows, cols, out);

  // 3) out = relu(AH @ W^T) in place ; 4 M-tiles per block
  pgcn_gemm_relu<<<(M_TILES + 3) / 4, 128, 0, stream>>>(W, out);
}